// RandomShear_46858093199724
// MI455X (gfx1250) — compile-verified
//
#include <hip/hip_runtime.h>
#include <cstdint>

#define IMG_H 512
#define IMG_W 512
#define IMG_C 3
#define ROWF  (IMG_W * IMG_C)   // 1536 floats = 6144 bytes per row
#define TPB   128               // 4 wave32 waves
#define SHEAR_RANGE 0.2f

// ---- Prologue: per-image tan(clip(angle)*0.2), computed once (B threads total)
__global__ void RandomShear_tan_kernel(const float* __restrict__ angles,
                                       float* __restrict__ tan_out, int B) {
    int i = blockIdx.x * blockDim.x + threadIdx.x;
    if (i < B) {
        float a = angles[i];
        a = fminf(fmaxf(a, -1.0f), 1.0f) * SHEAR_RANGE;
        tan_out[i] = tanf(a);
    }
}

__global__ __launch_bounds__(TPB) void RandomShear_46858093199724_kernel(
    const float* __restrict__ in,       // [B, 512, 512, 3]
    const float* __restrict__ tan_ang,  // [B] precomputed tan
    float* __restrict__ out)            // [B, 512, 512, 3]
{
    __shared__ float s_row[ROWF];

    const int y = blockIdx.x;          // row
    const int b = blockIdx.y;          // image
    const int t = threadIdx.x;

    const float* rowIn  = in  + ((size_t)b * IMG_H + (size_t)y) * ROWF;
    float*       rowOut = out + ((size_t)b * IMG_H + (size_t)y) * ROWF;

    // ---- Stage the input row into LDS via gfx1250 async global->LDS copies.
    // 3 iterations x 128 lanes x 16B = 6144B = one full row. Tracked by ASYNCcnt.
    #pragma unroll
    for (int k = 0; k < 3; ++k) {
        uint32_t f     = (uint32_t)(k * TPB + t) * 4u;            // float index
        uint32_t ldsB  = (uint32_t)(uintptr_t)(&s_row[f]);        // LDS byte addr
        uint32_t gofB  = f * 4u;                                  // global byte offset
        asm volatile("global_load_async_to_lds_b128 %0, %1, %2"
                     :: "v"(ldsB), "v"(gofB), "s"(rowIn)
                     : "memory");
    }

    // Uniform per-row shift: one scalar load + one multiply (no tanf here).
    const float shift = tan_ang[b] * (float)y;

    asm volatile("s_wait_asynccnt 0" ::: "memory");
    __syncthreads();

    // ---- 1-D lerp out of LDS, 4 consecutive floats per lane -> b128 stores.
    #pragma unroll
    for (int k = 0; k < 3; ++k) {
        const uint32_t base_f = (uint32_t)(k * TPB + t) * 4u;
        float r[4];
        #pragma unroll
        for (int j = 0; j < 4; ++j) {
            uint32_t f = base_f + (uint32_t)j;
            uint32_t x = f / 3u;                 // pixel  (magic-mul div)
            uint32_t c = f - 3u * x;             // channel

            float xin = (float)x + shift;
            float xf  = floorf(xin);
            float wx  = xin - xf;
            int   x0  = (int)xf;
            int   x1  = x0 + 1;

            int i0 = min(max(x0, 0), IMG_W - 1);
            int i1 = min(max(x1, 0), IMG_W - 1);
            float v0 = s_row[(uint32_t)i0 * 3u + c];
            float v1 = s_row[(uint32_t)i1 * 3u + c];
            v0 = (x0 >= 0 && x0 < IMG_W) ? v0 : 0.0f;   // CONSTANT fill = 0
            v1 = (x1 >= 0 && x1 < IMG_W) ? v1 : 0.0f;

            r[j] = v0 * (1.0f - wx) + v1 * wx;
        }
        *(float4*)(rowOut + base_f) = make_float4(r[0], r[1], r[2], r[3]);
    }
}

extern "C" void kernel_launch(void* const* d_in, const int* in_sizes, int n_in,
                              void* d_out, int out_size, void* d_ws, size_t ws_size,
                              hipStream_t stream) {
    const float* in     = (const float*)d_in[0];   // [B,512,512,3] fp32
    const float* angles = (const float*)d_in[1];   // [B] fp32
    float*       outp   = (float*)d_out;
    float*       tanw   = (float*)d_ws;            // B floats of scratch

    const int B = (n_in > 1) ? in_sizes[1] : 32;   // batch from angles' length

    RandomShear_tan_kernel<<<(B + 63) / 64, 64, 0, stream>>>(angles, tanw, B);

    dim3 grid(IMG_H, B, 1);
    dim3 block(TPB, 1, 1);
    RandomShear_46858093199724_kernel<<<grid, block, 0, stream>>>(in, tanw, outp);
}